// geodesic_loss_4930622456056
// MI455X (gfx1250) — compile-verified
//
#include <hip/hip_runtime.h>
#include <hip/hip_bf16.h>

// Geodesic rotation loss, B = 4,194,304 samples of 3x3 f32 matrices.
// Bandwidth-bound streaming reduce: 302 MB @ 23.3 TB/s => ~13us floor.
// Pass 1: grid-stride stream, 4 samples (144 B = 9 aligned float4) per
//         thread-iteration, NT-hinted b128 loads, per-block partial sums.
// Pass 2: single block reduces block partials, writes mean.
// Wave-level reduction uses V_WMMA_F32_16X16X4_F32 with an all-ones B
// matrix (layout-invariant, exact f32, deterministic rounding order).

typedef __attribute__((ext_vector_type(2))) float v2f;
typedef __attribute__((ext_vector_type(4))) float f4;
typedef __attribute__((ext_vector_type(8))) float v8f;

#define GEO_EPS 1e-6f

// Sum of one f32 value per lane across the full wave32.
// A (16x4): lane L holds A-row (L%16) entries {x, 0} (K split across lane
// halves per ISA layout). B (4x16) = all ones, so D[m][n] = rowsum(m) for
// every n regardless of B's exact lane layout.
// D layout: VGPR v, lanes 0-15 -> D[v][lane]; lanes 16-31 -> D[v+8][lane-16].
// So per-lane sum of the 8 D elements gives rowsums 0..7 (lanes 0-15) or
// rowsums 8..15 (lanes 16-31); one shfl_xor(16) combines the halves.
__device__ __forceinline__ float wave_sum_wmma(float x) {
    v2f a; a.x = x;    a.y = 0.0f;
    v2f b; b.x = 1.0f; b.y = 1.0f;
    v8f c = {};
    v8f d = __builtin_amdgcn_wmma_f32_16x16x4_f32(
        /*neg_a=*/false, a, /*neg_b=*/false, b,
        /*c_mod=*/(short)0, c, /*reuse_a=*/false, /*reuse_b=*/false);
    float s = ((d[0] + d[1]) + (d[2] + d[3])) + ((d[4] + d[5]) + (d[6] + d[7]));
    s += __shfl_xor(s, 16, 32);
    return s;
}

__device__ __forceinline__ float block_sum(float s) {
    __shared__ float wsum[8];
    float w = wave_sum_wmma(s);
    const int lane = threadIdx.x & 31;
    const int wid  = threadIdx.x >> 5;
    if (lane == 0) wsum[wid] = w;
    __syncthreads();
    float t = 0.0f;
    if (threadIdx.x == 0) {
#pragma unroll
        for (int i = 0; i < 8; ++i) t += wsum[i];
    }
    return t; // valid in thread 0 only
}

__global__ void geo_partial_kernel(const float* __restrict__ yp,
                                   const float* __restrict__ yt,
                                   float* __restrict__ partial,
                                   int ngroups /* = B/4 */) {
    float s = 0.0f;
    const int stride = gridDim.x * blockDim.x;
    for (int g = blockIdx.x * blockDim.x + threadIdx.x; g < ngroups; g += stride) {
        const f4* p4 = (const f4*)(yp + (size_t)g * 36u);
        const f4* t4 = (const f4*)(yt + (size_t)g * 36u);
        float prod[36];
#pragma unroll
        for (int i = 0; i < 9; ++i) {
            f4 p = __builtin_nontemporal_load(&p4[i]);   // global_load_b128 th:NT
            f4 t = __builtin_nontemporal_load(&t4[i]);
            prod[4 * i + 0] = p.x * t.x;
            prod[4 * i + 1] = p.y * t.y;
            prod[4 * i + 2] = p.z * t.z;
            prod[4 * i + 3] = p.w * t.w;
        }
#pragma unroll
        for (int k = 0; k < 4; ++k) {
            float dot = 0.0f;
#pragma unroll
            for (int j = 0; j < 9; ++j) dot += prod[9 * k + j];
            float cosv = fminf(fmaxf((dot - 1.0f) * 0.5f, -1.0f + GEO_EPS),
                               1.0f - GEO_EPS);
            s += acosf(cosv);
        }
    }
    float t = block_sum(s);
    if (threadIdx.x == 0) partial[blockIdx.x] = t;
}

__global__ void geo_final_kernel(const float* __restrict__ partial, int n,
                                 float* __restrict__ out, float inv_b) {
    float s = 0.0f;
    for (int i = threadIdx.x; i < n; i += blockDim.x) s += partial[i];
    float t = block_sum(s);
    if (threadIdx.x == 0) out[0] = t * inv_b;
}

extern "C" void kernel_launch(void* const* d_in, const int* in_sizes, int n_in,
                              void* d_out, int out_size, void* d_ws, size_t ws_size,
                              hipStream_t stream) {
    const float* ypred = (const float*)d_in[0];
    const float* ytrue = (const float*)d_in[1];
    float* out = (float*)d_out;
    float* partial = (float*)d_ws;

    const int total   = in_sizes[0];      // B * 9
    const int nsamp   = total / 9;        // B
    const int ngroups = total / 36;       // B / 4 (B is a multiple of 4)

    const int THREADS = 256;              // 8 wave32 per block
    const int BLOCKS  = 3072;             // ~24K waves in flight; 12 KB of ws

    geo_partial_kernel<<<BLOCKS, THREADS, 0, stream>>>(ypred, ytrue, partial, ngroups);
    geo_final_kernel<<<1, THREADS, 0, stream>>>(partial, BLOCKS, out,
                                                1.0f / (float)nsamp);
}